// KernelConv_11854109736983
// MI455X (gfx1250) — compile-verified
//
#include <hip/hip_runtime.h>

// KPN softmax-filter for MI455X (gfx1250).
// frames: [1,1,3,720,1280] f32 ; core: [1,147,720,1280] f32 ; out: [1,3,720,1280] f32
// Bandwidth-bound: 542MB single-use stream of `core` -> single pass, NT b128 loads,
// 4 px/thread. CDNA5 async global->LDS staging of the frame halo tile (ASYNCcnt).

typedef float v4f __attribute__((ext_vector_type(4)));

namespace {
constexpr int K    = 7;
constexpr int PADR = 3;
constexpr int Hh   = 720;
constexpr int Ww   = 1280;
constexpr int Cc   = 3;
constexpr int HW   = Hh * Ww;

constexpr int BX   = 16;               // threads x
constexpr int BY   = 16;               // threads y
constexpr int VEC  = 4;                // pixels per thread along w (b128)
constexpr int TWPX = BX * VEC;         // 64-px tile width
constexpr int THPX = BY;               // 16-px tile height
constexpr int SW   = TWPX + 2 * PADR;  // 70: staged tile width
constexpr int SH   = THPX + 2 * PADR;  // 22: staged tile height
constexpr int LDSW = 73;               // stride: +73 == +9 (mod 64) -> conflict-free
}

typedef __attribute__((address_space(1))) int glob_i32;
typedef __attribute__((address_space(3))) int lds_i32;

__global__ __launch_bounds__(256) void kpn_softmax_filter_gfx1250(
    const float* __restrict__ frames,
    const float* __restrict__ core,
    float* __restrict__ out)
{
    __shared__ float smem[SH * LDSW];   // 22*73*4 = 6424 B

    const int tx  = threadIdx.x;
    const int ty  = threadIdx.y;
    const int tid = ty * BX + tx;
    const int c   = blockIdx.z;
    const int h0  = blockIdx.y * THPX - PADR;   // halo-tile origin (frame coords)
    const int w0  = blockIdx.x * TWPX - PADR;

    const float* fplane = frames + (size_t)c * HW;

    // ---- Stage 70x22 halo tile of this channel's frame plane into LDS ----
    // CDNA5 async DMA (global_load_async_to_lds_b32, ASYNCcnt); zero-fill the
    // out-of-image halo with plain ds stores (disjoint addresses).
    for (int idx = tid; idx < SH * SW; idx += BX * BY) {
        const int r  = idx / SW;
        const int q  = idx - r * SW;
        const int gh = h0 + r;
        const int gw = w0 + q;
        const int l  = r * LDSW + q;
        if ((unsigned)gh < (unsigned)Hh && (unsigned)gw < (unsigned)Ww) {
            __builtin_amdgcn_global_load_async_to_lds_b32(
                (glob_i32*)(fplane + (size_t)gh * Ww + gw),
                (lds_i32*)(&smem[l]),
                /*offset=*/0, /*cpol=*/0);
        } else {
            smem[l] = 0.0f;
        }
    }
    asm volatile("s_wait_asynccnt 0" ::: "memory");  // my wave's DMAs done
    __syncthreads();                                  // everyone's DMAs done

    const int h  = blockIdx.y * THPX + ty;           // in range (720 = 45*16)
    const int wb = blockIdx.x * TWPX + tx * VEC;     // in range (1280 = 20*64)

    // core tap t for channel c lives in plane (t*3 + c): stride 3*HW elements.
    const float* cbase = core + (size_t)c * HW + (size_t)h * Ww + wb;

    float s0 = 0.f, s1 = 0.f, s2 = 0.f, s3 = 0.f;
    float a0 = 0.f, a1 = 0.f, a2 = 0.f, a3 = 0.f;

#pragma unroll
    for (int i = 0; i < K; ++i) {
        // Load the 10 halo values this thread's 4 pixels touch in row i once.
        float r[K + VEC - 1];
        const int rowbase = (ty + i) * LDSW + tx * VEC;
#pragma unroll
        for (int k = 0; k < K + VEC - 1; ++k)
            r[k] = smem[rowbase + k];

#pragma unroll
        for (int j = 0; j < K; ++j) {
            const int t = i * K + j;
            // Non-temporal b128: single-use 542MB stream, keep L2 for frames.
            const v4f x = __builtin_nontemporal_load(
                (const v4f*)(cbase + (size_t)t * (3 * HW)));
            const float e0 = __expf(x.x);
            const float e1 = __expf(x.y);
            const float e2 = __expf(x.z);
            const float e3 = __expf(x.w);
            s0 += e0;            s1 += e1;
            s2 += e2;            s3 += e3;
            a0 += e0 * r[j + 0]; a1 += e1 * r[j + 1];
            a2 += e2 * r[j + 2]; a3 += e3 * r[j + 3];
        }
    }

    v4f res;
    res.x = a0 / s0;
    res.y = a1 / s1;
    res.z = a2 / s2;
    res.w = a3 / s3;
    *(v4f*)(out + (size_t)c * HW + (size_t)h * Ww + wb) = res;
}

extern "C" void kernel_launch(void* const* d_in, const int* in_sizes, int n_in,
                              void* d_out, int out_size, void* d_ws, size_t ws_size,
                              hipStream_t stream) {
    const float* frames = (const float*)d_in[0];
    const float* core   = (const float*)d_in[1];
    float*       outp   = (float*)d_out;

    dim3 block(BX, BY, 1);                    // 256 threads = 8 wave32
    dim3 grid(Ww / TWPX, Hh / THPX, Cc);      // 20 x 45 x 3
    hipLaunchKernelGGL(kpn_softmax_filter_gfx1250, grid, block, 0, stream,
                       frames, core, outp);
}